// Gate_4277787427610
// MI455X (gfx1250) — compile-verified
//
#include <hip/hip_runtime.h>
#include <math.h>

// Problem constants (from the reference): B=8192, NUM=8, D=4096, fp32.
#define B_ROWS 8192
#define NUM_E  8
#define DIM    4096
#define KSPLIT 4
#define KCHUNK (DIM / KSPLIT)   // 1024 K-elements per wave

typedef float v2f __attribute__((ext_vector_type(2)));
typedef float v4f __attribute__((ext_vector_type(4)));
typedef float v8f __attribute__((ext_vector_type(8)));

// ---------------------------------------------------------------------------
// Kernel 1: partial gate logits via V_WMMA_F32_16X16X4_F32, K split 4 ways.
//
// Wave w: row tile = w>>2 (16 rows), k-chunk = w&3 (1024 K). 2048 waves total.
// Partial logits go to part[row][kc][n] (fp32, fixed order -> deterministic).
//
// VGPR layouts (CDNA5 ISA 7.12.2, 32-bit data):
//   A (16x4, MxK):  lane L: M = L&15;  VGPR j holds K = j + 2*(L>>4)
//   B (4x16, KxN):  lane L: N = L&15;  VGPR j holds K = j + 2*(L>>4)
//   => per lane both fragments are one contiguous float2 (global_load_b64).
//   C/D (16x16):    VGPR r: lanes 0-15: M=r, N=lane; lanes 16-31: M=r+8.
//
// B columns N=8..15 are fed with duplicates of N-8 (no masking needed): the
// resulting C columns are finite garbage that is simply never stored, and
// column results never mix across lanes. EXEC stays all-ones throughout.
// ---------------------------------------------------------------------------
__global__ __launch_bounds__(128) void gate_partial_wmma_kernel(
    const float* __restrict__ x,    // [B, D]
    const float* __restrict__ W,    // [NUM, D]
    float* __restrict__ part)       // [B, KSPLIT, NUM] workspace
{
    const int lane    = threadIdx.x & 31;
    const int wave    = (blockIdx.x * blockDim.x + threadIdx.x) >> 5;
    const int tile    = wave >> 2;           // 512 row tiles
    const int kc      = wave & (KSPLIT - 1); // K chunk
    const int rowBase = tile * 16;
    const int half    = lane >> 4;           // 0 -> K{0,1}, 1 -> K{2,3}
    const int nl      = lane & 15;           // A: row-in-tile; B/C: column N
    const int k0      = kc * KCHUNK;

    const float* __restrict__ ap =
        x + (size_t)(rowBase + nl) * DIM + k0 + 2 * half;
    const float* __restrict__ bp =
        W + (size_t)(nl & (NUM_E - 1)) * DIM + k0 + 2 * half;

    v8f c = {0.f, 0.f, 0.f, 0.f, 0.f, 0.f, 0.f, 0.f};

#pragma unroll 16
    for (int k = 0; k < KCHUNK; k += 4) {
        v2f a = *(const v2f*)(ap + k);
        v2f b = *(const v2f*)(bp + k);
        // 8 args: (neg_a, A, neg_b, B, c_mod, C, reuse_a, reuse_b)
        c = __builtin_amdgcn_wmma_f32_16x16x4_f32(
            false, a, false, b, (short)0, c, false, false);
    }

    // Store the 8 valid columns; lane holds rows M = r + 8*half for r=0..7.
    if (nl < NUM_E) {
#pragma unroll
        for (int r = 0; r < 8; ++r) {
            const int row = rowBase + r + 8 * half;
            part[(size_t)row * (KSPLIT * NUM_E) + kc * NUM_E + nl] = c[r];
        }
    }
}

// ---------------------------------------------------------------------------
// Kernel 2: fused softmax + weighted combine.
// One block per row. Each thread reduces the row's 4x8 uniform partial logits
// (scalarizable loads) and computes the 8-wide softmax redundantly in
// registers (8 v_exp_f32 -- noise next to 128 KB of streaming per block),
// then streams experts with coalesced non-temporal float4 accesses
// (1.07 GB read + 134 MB write, touched once, >> 192 MB L2).
// ---------------------------------------------------------------------------
__global__ __launch_bounds__(256) void softmax_combine_kernel(
    const float* __restrict__ experts,  // [B, NUM, D]
    const float* __restrict__ part,     // [B, KSPLIT, NUM]
    float* __restrict__ out)            // [B, D]
{
    const int b = blockIdx.x;

    // Reduce K-split partials -> logits (fixed order, deterministic).
    const float* __restrict__ pb = part + (size_t)b * (KSPLIT * NUM_E);
    float lg[NUM_E];
#pragma unroll
    for (int n = 0; n < NUM_E; ++n)
        lg[n] = (pb[n] + pb[NUM_E + n]) + (pb[2 * NUM_E + n] + pb[3 * NUM_E + n]);

    // 8-wide softmax in registers.
    float mx = lg[0];
#pragma unroll
    for (int n = 1; n < NUM_E; ++n) mx = fmaxf(mx, lg[n]);
    float wgt[NUM_E];
    float s = 0.f;
#pragma unroll
    for (int n = 0; n < NUM_E; ++n) {
        wgt[n] = __expf(lg[n] - mx);
        s += wgt[n];
    }
    const float inv = __fdividef(1.f, s);
#pragma unroll
    for (int n = 0; n < NUM_E; ++n) wgt[n] *= inv;

    // Bandwidth phase: out[b,:] = sum_n wgt[n] * experts[b,n,:]
    const v4f* __restrict__ eb = (const v4f*)(experts + (size_t)b * NUM_E * DIM);
    v4f* __restrict__ ob = (v4f*)(out + (size_t)b * DIM);

    constexpr int C4 = DIM / 4;  // 1024 float4 chunks per row
#pragma unroll
    for (int cidx = threadIdx.x; cidx < C4; cidx += 256) {
        v4f acc = {0.f, 0.f, 0.f, 0.f};
#pragma unroll
        for (int nn = 0; nn < NUM_E; ++nn) {
            v4f e = __builtin_nontemporal_load(eb + (size_t)nn * C4 + cidx);
            acc += e * wgt[nn];
        }
        __builtin_nontemporal_store(acc, ob + cidx);
    }
}

// ---------------------------------------------------------------------------
extern "C" void kernel_launch(void* const* d_in, const int* in_sizes, int n_in,
                              void* d_out, int out_size, void* d_ws, size_t ws_size,
                              hipStream_t stream) {
    const float* x       = (const float*)d_in[0];  // [B, D]
    const float* experts = (const float*)d_in[1];  // [B, NUM, D]
    const float* W       = (const float*)d_in[2];  // [NUM, D]
    float* out  = (float*)d_out;                   // [B, D]
    float* part = (float*)d_ws;                    // [B, KSPLIT, NUM] = 1 MB

    // 512 row tiles x 4 K-chunks = 2048 waves / 4 waves per block = 512 blocks.
    gate_partial_wmma_kernel<<<dim3(B_ROWS / 16 * KSPLIT / 4), dim3(128), 0,
                               stream>>>(x, W, part);
    // One block per row; 8192 blocks x 8 waves for the bandwidth-bound stream.
    softmax_combine_kernel<<<dim3(B_ROWS), dim3(256), 0, stream>>>(experts, part,
                                                                   out);
}